// SQEmbedding_35485019800073
// MI455X (gfx1250) — compile-verified
//
#include <hip/hip_runtime.h>
#include <hip/hip_bf16.h>
#include <math.h>

typedef float v2f __attribute__((ext_vector_type(2)));
typedef float v8f __attribute__((ext_vector_type(8)));

#define B_  8
#define T_  4096
#define D_  64
#define M_  1024
#define N_  (B_*T_)

#define ROWS 16
#define TPB  128
#define MCH  64
#define XS_LD 68
#define EC_LD 68
#define SC_LD 1028

// workspace layout (float slots)
#define WS_EH     0
#define WS_CNT    1024
#define WS_RECON  2048
#define WS_NEGENT 2056

// ---------------------------------------------------------------------------
// Init: eh[m] = 0.5*precision*||e_m||^2 ; zero counters/accumulators each call
// ---------------------------------------------------------------------------
__global__ __launch_bounds__(256) void vq_init(const float* __restrict__ emb,
                                               const float* __restrict__ lvq,
                                               float* __restrict__ ws) {
  int m = blockIdx.x * blockDim.x + threadIdx.x;
  if (m >= M_) return;
  float prec = __expf(-lvq[0]);
  float s = 0.f;
  const float4* e4 = (const float4*)(emb + m * D_);
  #pragma unroll
  for (int i = 0; i < D_/4; ++i) {
    float4 v = e4[i];
    s += v.x*v.x + v.y*v.y + v.z*v.z + v.w*v.w;
  }
  ws[WS_EH + m] = 0.5f * prec * s;
  ((unsigned*)ws)[WS_CNT + m] = 0u;
  if (m < B_) { ws[WS_RECON + m] = 0.f; ws[WS_NEGENT + m] = 0.f; }
}

// ---------------------------------------------------------------------------
// Main fused kernel: one WG (4 waves, 128 thr) handles 16 rows of x_flat.
// Pass A: scores z = prec*(x.e) - eh via WMMA; t=(z+g)/temp -> LDS; online
//         logsumexp/argmax stats. Pass B: quantized = softmax(t) @ E via WMMA.
// ---------------------------------------------------------------------------
__global__ __launch_bounds__(TPB) void vq_main(const float* __restrict__ x,
    const float* __restrict__ emb, const float* __restrict__ lvq,
    const float* __restrict__ gum, const float* __restrict__ tmp,
    float* __restrict__ outq, float* __restrict__ ws) {
  extern __shared__ float smem[];
  float* xs    = smem;                  // ROWS * XS_LD  (x tile, later q tile)
  float* ec    = xs + ROWS*XS_LD;       // MCH * EC_LD   (embedding chunk)
  float* sc    = ec + MCH*EC_LD;        // ROWS * SC_LD  (t scores, fp32)
  float* pst   = sc + ROWS*SC_LD;       // ROWS*4*8      (per-wave stat scratch)
  float* rowmt = pst + ROWS*4*8;        // 16
  float* rowis = rowmt + ROWS;          // 16
  float* redp  = rowis + ROWS;          // 1

  const int tid  = threadIdx.x;
  const int lane = tid & 31;
  const int wv   = tid >> 5;
  const int lc   = lane & 15;           // lane column within 16
  const int h    = lane >> 4;           // lane half
  const int koff = h ? 2 : 0;

  const int n0 = blockIdx.x * ROWS;
  const int b  = n0 >> 12;              // / T_
  const int t0 = n0 & (T_ - 1);
  const long xbase = (long)b * D_ * T_ + t0;

  const float prec  = __expf(-lvq[0]);
  const float itemp = 1.0f / tmp[0];

  if (tid == 0) redp[0] = 0.f;

  // stage x tile (B,D,T layout -> xs[r][d])
  #pragma unroll
  for (int i = 0; i < 8; ++i) {
    int idx = tid + i*TPB;
    int d = idx >> 4, r = idx & 15;
    xs[r*XS_LD + d] = x[xbase + (long)d*T_ + r];
  }
  __syncthreads();

  // preload A fragments (x rows, 16 K-steps of 4)
  v2f af[16];
  #pragma unroll
  for (int kk = 0; kk < 16; ++kk) {
    af[kk].x = xs[lc*XS_LD + kk*4 + koff];
    af[kk].y = xs[lc*XS_LD + kk*4 + koff + 1];
  }

  // per-lane online stats for rows j+8h over this lane's column subset
  float mz[8], idv[8], Se[8], Ze[8], mt[8], St[8];
  #pragma unroll
  for (int j = 0; j < 8; ++j) {
    mz[j] = -1e30f; idv[j] = 0.f; Se[j] = 0.f; Ze[j] = 0.f;
    mt[j] = -1e30f; St[j] = 0.f;
  }

  // ---------------- Pass A ----------------
  for (int m0 = 0; m0 < M_; m0 += MCH) {
    __syncthreads();
    #pragma unroll
    for (int i = 0; i < 8; ++i) {            // stage E chunk (64 x 64 f32)
      int idx = tid + i*TPB;
      int mm = idx >> 4, d4 = idx & 15;
      float4 v = ((const float4*)emb)[(m0 + mm)*(D_/4) + d4];
      *(float4*)&ec[mm*EC_LD + d4*4] = v;
    }
    __syncthreads();

    const int mloc = wv*16 + lc;             // this wave's 16 codebook cols
    const int mcol = m0 + mloc;

    v8f acc = {0.f,0.f,0.f,0.f,0.f,0.f,0.f,0.f};
    #pragma unroll
    for (int kk = 0; kk < 16; ++kk) {        // K = 64 in steps of 4
      v2f bb;
      bb.x = ec[mloc*EC_LD + kk*4 + koff];
      bb.y = ec[mloc*EC_LD + kk*4 + koff + 1];
      acc = __builtin_amdgcn_wmma_f32_16x16x4_f32(
              false, af[kk], false, bb, (short)0, acc, false, false);
    }

    float ehv = ws[WS_EH + mcol];
    #pragma unroll
    for (int j = 0; j < 8; ++j) {
      int r = j + 8*h;
      float g = gum[(long)(n0 + r)*M_ + mcol];  // stream gumbels exactly once
      float z = prec*acc[j] - ehv;              // logit up to row-constant
      float t = (z + g)*itemp;
      sc[r*SC_LD + mcol] = t;
      // plain-softmax entropy stats + argmax (argmin distance)
      float nm = fmaxf(mz[j], z);
      float eo = __expf(mz[j]-nm), en = __expf(z-nm);
      Se[j] = Se[j]*eo + en;
      Ze[j] = Ze[j]*eo + z*en;
      idv[j] = (z > mz[j]) ? (float)mcol : idv[j];
      mz[j] = nm;
      // gumbel-softmax denominator stats
      float nmt = fmaxf(mt[j], t);
      St[j] = St[j]*__expf(mt[j]-nmt) + __expf(t-nmt);
      mt[j] = nmt;
    }
  }

  // butterfly-reduce stats across the 16 lane-columns (halves stay separate)
  #pragma unroll
  for (int msk = 1; msk < 16; msk <<= 1) {
    #pragma unroll
    for (int j = 0; j < 8; ++j) {
      float omz = __shfl_xor(mz[j],  msk, 32);
      float oid = __shfl_xor(idv[j], msk, 32);
      float oS  = __shfl_xor(Se[j],  msk, 32);
      float oZ  = __shfl_xor(Ze[j],  msk, 32);
      float omt = __shfl_xor(mt[j],  msk, 32);
      float oSt = __shfl_xor(St[j],  msk, 32);
      float nm = fmaxf(mz[j], omz);
      float e0 = __expf(mz[j]-nm), e1 = __expf(omz-nm);
      Se[j] = Se[j]*e0 + oS*e1;
      Ze[j] = Ze[j]*e0 + oZ*e1;
      idv[j] = (omz > mz[j]) ? oid : idv[j];
      mz[j] = nm;
      float nmt = fmaxf(mt[j], omt);
      St[j] = St[j]*__expf(mt[j]-nmt) + oSt*__expf(omt-nmt);
      mt[j] = nmt;
    }
  }
  if (lc == 0) {
    #pragma unroll
    for (int j = 0; j < 8; ++j) {
      int r = j + 8*h;
      float* p = &pst[(r*4 + wv)*8];
      p[0]=mz[j]; p[1]=idv[j]; p[2]=Se[j]; p[3]=Ze[j]; p[4]=mt[j]; p[5]=St[j];
    }
  }
  __syncthreads();

  if (tid < ROWS) {                       // merge the 4 waves' partials per row
    int r = tid;
    float* p = &pst[(r*4 + 0)*8];
    float fmz=p[0], fid=p[1], fS=p[2], fZ=p[3], fmt=p[4], fSt=p[5];
    for (int w = 1; w < 4; ++w) {
      float* q = &pst[(r*4 + w)*8];
      float nm = fmaxf(fmz, q[0]);
      float e0 = __expf(fmz-nm), e1 = __expf(q[0]-nm);
      fS = fS*e0 + q[2]*e1;
      fZ = fZ*e0 + q[3]*e1;
      fid = (q[0] > fmz) ? q[1] : fid;
      fmz = nm;
      float nmt = fmaxf(fmt, q[4]);
      fSt = fSt*__expf(fmt-nmt) + q[5]*__expf(q[4]-nmt);
      fmt = nmt;
    }
    float negent = fZ/fS - fmz - __logf(fS);   // sum_m p*(logp)
    atomicAdd(&ws[WS_NEGENT + b], negent);
    atomicAdd(&((unsigned*)ws)[WS_CNT + (int)fid], 1u);
    rowmt[r] = fmt;
    rowis[r] = 1.0f / fSt;
  }
  __syncthreads();

  const float mtl = rowmt[lc];
  float invs[8];
  #pragma unroll
  for (int j = 0; j < 8; ++j) invs[j] = rowis[j + 8*h];

  // ---------------- Pass B: quantized = softmax(t) @ E ----------------
  const int dcol = wv*16 + lc;            // this wave's 16 output dims
  v8f accq = {0.f,0.f,0.f,0.f,0.f,0.f,0.f,0.f};
  for (int m0 = 0; m0 < M_; m0 += MCH) {
    __syncthreads();
    #pragma unroll
    for (int i = 0; i < 8; ++i) {         // re-stage E chunk (L2-resident)
      int idx = tid + i*TPB;
      int mm = idx >> 4, d4 = idx & 15;
      float4 v = ((const float4*)emb)[(m0 + mm)*(D_/4) + d4];
      *(float4*)&ec[mm*EC_LD + d4*4] = v;
    }
    __syncthreads();
    #pragma unroll
    for (int kk = 0; kk < 16; ++kk) {     // K = 64 codes per chunk
      int ke = kk*4 + koff;
      v2f a;
      a.x = __expf(sc[lc*SC_LD + m0 + ke]     - mtl);
      a.y = __expf(sc[lc*SC_LD + m0 + ke + 1] - mtl);
      v2f bb;
      bb.x = ec[ke*EC_LD + dcol];
      bb.y = ec[(ke+1)*EC_LD + dcol];
      accq = __builtin_amdgcn_wmma_f32_16x16x4_f32(
               false, a, false, bb, (short)0, accq, false, false);
    }
  }

  // normalize, recon accumulation, transpose-store via LDS
  float rsum = 0.f;
  float qv[8];
  #pragma unroll
  for (int j = 0; j < 8; ++j) {
    float q  = accq[j]*invs[j];
    float xv = xs[(j + 8*h)*XS_LD + dcol];
    float df = xv - q;
    rsum += df*df;
    qv[j] = q;
  }
  atomicAdd(redp, rsum);
  __syncthreads();
  #pragma unroll
  for (int j = 0; j < 8; ++j) xs[(j + 8*h)*XS_LD + dcol] = qv[j];
  if (tid == 0) atomicAdd(&ws[WS_RECON + b], redp[0]);
  __syncthreads();
  #pragma unroll
  for (int i = 0; i < 8; ++i) {           // coalesced (B,D,T) store
    int idx = tid + i*TPB;
    int d = idx >> 4, r = idx & 15;
    outq[xbase + (long)d*T_ + r] = xs[r*XS_LD + d];
  }
}

// ---------------------------------------------------------------------------
// Final: perplexity from histogram + loss = mean(recon + neg_entropy)
// ---------------------------------------------------------------------------
__global__ __launch_bounds__(1024) void vq_final(const float* __restrict__ lvq,
                                                 const float* __restrict__ ws,
                                                 float* __restrict__ out) {
  __shared__ float red[1024];
  int t = threadIdx.x;
  unsigned c = ((const unsigned*)ws)[WS_CNT + t];
  float p = (float)c * (1.0f/(float)N_);
  red[t] = p * __logf(p + 1e-10f);
  __syncthreads();
  for (int s = 512; s > 0; s >>= 1) {
    if (t < s) red[t] += red[t + s];
    __syncthreads();
  }
  if (t == 0) {
    float prec = __expf(-lvq[0]);
    float loss = 0.f;
    for (int bb = 0; bb < B_; ++bb)
      loss += 0.5f*prec*ws[WS_RECON + bb] + ws[WS_NEGENT + bb];
    out[(long)B_*D_*T_ + 0] = loss / (float)B_;
    out[(long)B_*D_*T_ + 1] = __expf(-red[0]);
  }
}

extern "C" void kernel_launch(void* const* d_in, const int* in_sizes, int n_in,
                              void* d_out, int out_size, void* d_ws, size_t ws_size,
                              hipStream_t stream) {
  const float* x   = (const float*)d_in[0];
  const float* emb = (const float*)d_in[1];
  const float* lvq = (const float*)d_in[2];
  const float* gum = (const float*)d_in[3];
  const float* tmp = (const float*)d_in[4];
  float* out = (float*)d_out;
  float* ws  = (float*)d_ws;

  vq_init<<<4, 256, 0, stream>>>(emb, lvq, ws);

  size_t shmem = (size_t)(ROWS*XS_LD + MCH*EC_LD + ROWS*SC_LD
                          + ROWS*4*8 + ROWS + ROWS + 1) * sizeof(float);
  vq_main<<<N_/ROWS, TPB, shmem, stream>>>(x, emb, lvq, gum, tmp, out, ws);

  vq_final<<<1, 1024, 0, stream>>>(lvq, ws, out);
}